// RNN_57904749085027
// MI455X (gfx1250) — compile-verified
//
#include <hip/hip_runtime.h>
#include <hip/hip_bf16.h>
#include <math.h>

// ---------------------------------------------------------------------------
// Problem constants (from reference)
// ---------------------------------------------------------------------------
#define HID    100
#define SEQ    480
#define INP    640
#define NCLS   307200
#define G4H    400        // 4*HID
#define NLAY   4
#define NSTG   3

typedef __attribute__((ext_vector_type(2))) float v2f;
typedef __attribute__((ext_vector_type(4))) float v4f;
typedef __attribute__((ext_vector_type(8))) float v8f;

// ---------------------------------------------------------------------------
// Kernel 1: xp = X * Wih^T + (bih + bhh), via V_WMMA_F32_16X16X4_F32.
//   X   : (SEQ, K)    row-major, K compile-time (640 or 100)
//   W   : (G4H, K)    row-major (so B[k][n] = W[n][k])
//   xp  : (SEQ, G4H)
// One wave per 32x16 (MxN) macro-tile: two M-tiles share each B fragment,
// so the K-step is 3 loads -> 2 WMMAs (B reused, halved W traffic).
// Fragment layouts per CDNA5 ISA 7.12.2:
//   A (16x4 f32): lanes 0-15 hold K=0,1 in v[0],v[1]; lanes 16-31 hold K=2,3.
//   B (4x16 f32): mirrored (lane%16 = N, lane/16 selects K pair).
//   C/D (16x16 f32): VGPR r -> M=r (lanes 0-15) / M=r+8 (lanes 16-31), N=lane%16.
// ---------------------------------------------------------------------------
template <int K>
__global__ __launch_bounds__(32)
void gemm_xp_kernel(const float* __restrict__ X,
                    const float* __restrict__ W,
                    const float* __restrict__ bih,
                    const float* __restrict__ bhh,
                    float* __restrict__ xp)
{
    const int lane  = threadIdx.x & 31;
    const int laneN = lane & 15;
    const int half  = lane >> 4;          // 0: K pair {0,1}, 1: K pair {2,3}
    const int m0    = blockIdx.y * 32;    // time macro-tile (two 16-row tiles)
    const int n0    = blockIdx.x * 16;    // gate tile

    // Fused bias: depends only on column N -> broadcast into all acc rows.
    const float bias = bih[n0 + laneN] + bhh[n0 + laneN];
    v8f c0, c1;
#pragma unroll
    for (int i = 0; i < 8; ++i) { c0[i] = bias; c1[i] = bias; }

    const float* __restrict__ Xr0  = X + (size_t)(m0      + laneN) * K; // A tile 0 row
    const float* __restrict__ Xr1  = X + (size_t)(m0 + 16 + laneN) * K; // A tile 1 row
    const float* __restrict__ Wrow = W + (size_t)(n0      + laneN) * K; // B col N

    const int kofs = half * 2;
#pragma unroll 4
    for (int k = 0; k < K; k += 4) {
        v2f a0, a1, b;
        b[0]  = Wrow[k + kofs + 0];
        b[1]  = Wrow[k + kofs + 1];
        a0[0] = Xr0[k + kofs + 0];
        a0[1] = Xr0[k + kofs + 1];
        a1[0] = Xr1[k + kofs + 0];
        a1[1] = Xr1[k + kofs + 1];
        // (neg_a, A, neg_b, B, c_mod, C, reuse_a, reuse_b)
        c0 = __builtin_amdgcn_wmma_f32_16x16x4_f32(false, a0, false, b,
                                                   (short)0, c0, false, false);
        c1 = __builtin_amdgcn_wmma_f32_16x16x4_f32(false, a1, false, b,
                                                   (short)0, c1, false, false);
    }

    float* __restrict__ o0 = xp + (size_t)(m0      + half * 8) * G4H + n0 + laneN;
    float* __restrict__ o1 = xp + (size_t)(m0 + 16 + half * 8) * G4H + n0 + laneN;
#pragma unroll
    for (int r = 0; r < 8; ++r) {
        o0[(size_t)r * G4H] = c0[r];
        o1[(size_t)r * G4H] = c1[r];
    }
}

// ---------------------------------------------------------------------------
// Kernel 2: sequential LSTM scan over SEQ steps (single workgroup).
//   xp  : (SEQ, G4H) precomputed input projections (bias folded in)
//   Whh : (G4H, HID)
//   hs  : (SEQ, HID) output hidden sequence
// Thread tid<400 owns gate row `tid`: its Whh row lives in 100 registers for
// the whole scan; h broadcast from LDS. Threads 0..99 own c[j] in a register.
// Gate order (PyTorch): rows 0-99=i, 100-199=f, 200-299=g, 300-399=o.
// ---------------------------------------------------------------------------
__global__ __launch_bounds__(512)
void lstm_scan_kernel(const float* __restrict__ xp,
                      const float* __restrict__ Whh,
                      float* __restrict__ hs)
{
    __shared__ float h_s[HID];
    __shared__ float g_s[G4H];

    const int tid = threadIdx.x;

    float w[HID];
    if (tid < G4H) {
        const float* __restrict__ wr = Whh + (size_t)tid * HID;
#pragma unroll
        for (int k = 0; k < HID; ++k) w[k] = wr[k];
    }

    float cst = 0.0f;
    if (tid < HID) h_s[tid] = 0.0f;
    __syncthreads();

    for (int t = 0; t < SEQ; ++t) {
        if (tid < G4H) {
            float acc = xp[(size_t)t * G4H + tid];
#pragma unroll
            for (int k = 0; k < HID; ++k) acc += w[k] * h_s[k];
            g_s[tid] = acc;
        }
        __syncthreads();
        if (tid < HID) {
            const float ig = g_s[tid];
            const float fg = g_s[HID + tid];
            const float gg = g_s[2 * HID + tid];
            const float og = g_s[3 * HID + tid];
            const float si = 1.0f / (1.0f + __expf(-ig));
            const float sf = 1.0f / (1.0f + __expf(-fg));
            const float so = 1.0f / (1.0f + __expf(-og));
            cst = sf * cst + si * tanhf(gg);
            const float hn = so * tanhf(cst);
            h_s[tid] = hn;
            hs[(size_t)t * HID + tid] = hn;
        }
        __syncthreads();
    }
}

// ---------------------------------------------------------------------------
// Kernel 3: FC head + optional blend. Memory-bound streamer over fcW.
//   out[r] = fcb[r] + dot(fcW[r, :], h);  out = (out + blend)*0.5 if blend.
// One thread per row; 25 x 16B loads per row (contiguous 400B).
// NT=false (stage 0): regular temporal loads -> 122.9MB slab can stay
//   resident in the 192MB L2 across graph replays.
// NT=true (stages 1-2): nontemporal loads stream through without evicting
//   stage 0's slab (all-RT would cyclically thrash 369MB through 192MB L2).
// ---------------------------------------------------------------------------
template <bool NT>
__global__ __launch_bounds__(256)
void fc_blend_kernel(const float* __restrict__ h,
                     const float* __restrict__ fcW,
                     const float* __restrict__ fcb,
                     const float* __restrict__ blend,   // may be nullptr
                     float* __restrict__ out)
{
    __shared__ float h_s[HID];
    const int tid = threadIdx.x;
    if (tid < HID) h_s[tid] = h[tid];
    __syncthreads();

    const size_t r = (size_t)blockIdx.x * 256 + tid;   // NCLS == 1200*256
    const v4f* __restrict__ wr = (const v4f*)(fcW + r * HID);
    float acc = fcb[r];
#pragma unroll
    for (int k4 = 0; k4 < HID / 4; ++k4) {
        const v4f v = NT ? __builtin_nontemporal_load(&wr[k4]) : wr[k4];
        acc += v[0] * h_s[4 * k4 + 0]
             + v[1] * h_s[4 * k4 + 1]
             + v[2] * h_s[4 * k4 + 2]
             + v[3] * h_s[4 * k4 + 3];
    }
    out[r] = blend ? (acc + blend[r]) * 0.5f : acc;
}

// ---------------------------------------------------------------------------
// Host orchestration
// ---------------------------------------------------------------------------
extern "C" void kernel_launch(void* const* d_in, const int* in_sizes, int n_in,
                              void* d_out, int out_size, void* d_ws, size_t ws_size,
                              hipStream_t stream)
{
    (void)in_sizes; (void)n_in; (void)out_size; (void)ws_size;

    const float* x    = (const float*)d_in[0];   // (1,480,640)
    const float* xn   = (const float*)d_in[1];
    const float* xnn  = (const float*)d_in[2];
    const float* Wih0 = (const float*)d_in[3];   // (3,400,640)
    const float* WihR = (const float*)d_in[4];   // (3,3,400,100)
    const float* Whh  = (const float*)d_in[5];   // (3,4,400,100)
    const float* bih  = (const float*)d_in[6];   // (3,4,400)
    const float* bhh  = (const float*)d_in[7];   // (3,4,400)
    const float* fcW  = (const float*)d_in[8];   // (3,307200,100)
    const float* fcb  = (const float*)d_in[9];   // (3,307200)
    float* out        = (float*)d_out;           // (1,480,640)

    // Workspace carve-up (floats): cur | xp | hsA | hsB   (~2.4 MB total)
    float* ws  = (float*)d_ws;
    float* cur = ws;                       // NCLS          = 307200
    float* xp  = cur + NCLS;               // SEQ*G4H       = 192000
    float* hsA = xp + (size_t)SEQ * G4H;   // SEQ*HID       =  48000
    float* hsB = hsA + (size_t)SEQ * HID;  // SEQ*HID       =  48000

    const dim3 gemm_grid(G4H / 16, SEQ / 32);   // 25 x 15 macro-tiles

    const float* stage_in = x;
    for (int s = 0; s < NSTG; ++s) {
        const float* hs_prev = nullptr;
        for (int l = 0; l < NLAY; ++l) {
            const float* bi = bih + ((size_t)s * NLAY + l) * G4H;
            const float* bh = bhh + ((size_t)s * NLAY + l) * G4H;

            if (l == 0) {
                const float* W = Wih0 + (size_t)s * G4H * INP;
                gemm_xp_kernel<INP><<<gemm_grid, 32, 0, stream>>>(
                    stage_in, W, bi, bh, xp);
            } else {
                const float* W =
                    WihR + ((size_t)s * (NLAY - 1) + (l - 1)) * G4H * HID;
                gemm_xp_kernel<HID><<<gemm_grid, 32, 0, stream>>>(
                    hs_prev, W, bi, bh, xp);
            }

            float* hs = (l & 1) ? hsB : hsA;
            const float* whh = Whh + ((size_t)s * NLAY + l) * G4H * HID;
            lstm_scan_kernel<<<1, 512, 0, stream>>>(xp, whh, hs);
            hs_prev = hs;
        }

        const float* hlast = hs_prev + (size_t)(SEQ - 1) * HID;
        const float* blend = (s == 0) ? xn : (s == 1) ? xnn : nullptr;
        float* dst         = (s == NSTG - 1) ? out : cur;
        const float* fw    = fcW + (size_t)s * NCLS * HID;
        const float* fb    = fcb + (size_t)s * NCLS;

        if (s == 0)
            fc_blend_kernel<false><<<NCLS / 256, 256, 0, stream>>>(
                hlast, fw, fb, blend, dst);
        else
            fc_blend_kernel<true><<<NCLS / 256, 256, 0, stream>>>(
                hlast, fw, fb, blend, dst);

        stage_in = cur;
    }
}